// FeatureRefinementGNN_31147102830709
// MI455X (gfx1250) — compile-verified
//
#include <hip/hip_runtime.h>

#define N_NODES 100000
#define N_EDGES 1600000
#define CH 128

typedef __attribute__((ext_vector_type(2))) float v2f;
typedef __attribute__((ext_vector_type(8))) float v8f;

// Hardware f32 atomic add (L2 atomic unit), device scope, no return (STOREcnt).
__device__ __forceinline__ void atomic_add_f32_hw(float* p, float v) {
  asm volatile("global_atomic_add_f32 %0, %1, off scope:SCOPE_DEV"
               :: "v"(p), "v"(v) : "memory");
}

// Async DMA: 16 bytes global -> LDS, tracked by ASYNCcnt (no VGPR round-trip).
__device__ __forceinline__ void async_copy_b128(void* lds_ptr, const void* gptr) {
  unsigned lds_off = (unsigned)(size_t)lds_ptr;  // low 32 bits = LDS offset
  asm volatile("global_load_async_to_lds_b128 %0, %1, off"
               :: "v"(lds_off), "v"(gptr) : "memory");
}

__device__ __forceinline__ void wait_async_all() {
  asm volatile("s_wait_asynccnt 0x0" ::: "memory");
}

// ---------------------------------------------------------------------------
// Zero a float buffer (grid-stride, float4).
// ---------------------------------------------------------------------------
__global__ void zero_kernel(float4* __restrict__ p, int n4) {
  int i = blockIdx.x * blockDim.x + threadIdx.x;
  int stride = gridDim.x * blockDim.x;
  const float4 z = make_float4(0.f, 0.f, 0.f, 0.f);
  for (; i < n4; i += stride) p[i] = z;
}

// ---------------------------------------------------------------------------
// In-degree counts: cnt[dst[e]] += 1.0f
// ---------------------------------------------------------------------------
__global__ void count_kernel(const int* __restrict__ dst, float* __restrict__ cnt) {
  int e = blockIdx.x * 256 + threadIdx.x;
  if (e < N_EDGES) atomic_add_f32_hw(&cnt[dst[e]], 1.0f);
}

// ---------------------------------------------------------------------------
// Edge scatter: one wave (32 lanes) per edge, each lane owns 4 channels.
// Coalesced float4 gather of h[src] + 4 hardware f32 atomics into agg[dst].
// agg buffer (51 MB) is L2-resident on MI455X (192 MB L2).
// ---------------------------------------------------------------------------
__global__ void scatter_kernel(const float* __restrict__ h, const int* __restrict__ src,
                               const int* __restrict__ dst, float* __restrict__ agg) {
  long long idx = (long long)blockIdx.x * 256 + threadIdx.x;
  int e = (int)(idx >> 5);
  int c = ((int)idx & 31) << 2;
  int s = src[e];
  int d = dst[e];
  const float4 v = *(const float4*)(h + (size_t)s * CH + c);
  float* out = agg + (size_t)d * CH + c;
  atomic_add_f32_hw(out + 0, v.x);
  atomic_add_f32_hw(out + 1, v.y);
  atomic_add_f32_hw(out + 2, v.z);
  atomic_add_f32_hw(out + 3, v.w);
}

// ---------------------------------------------------------------------------
// Fused SAGE layer: hout = relu( (agg/max(cnt,1)) @ Wl + b + hin @ Wr )
// Block = 256 threads = 8 waves; 16 nodes/block; wave w owns out cols
// [16w,16w+16). Raw A tiles DMA'd to LDS via global_load_async_to_lds_b128;
// mean division deferred to the epilogue (diag(inv)*(A@W) == (diag(inv)A)@W)
// using a second accumulator chain. K=128 swept with V_WMMA_F32_16X16X4_F32.
// ---------------------------------------------------------------------------
__launch_bounds__(256, 2)
__global__ void sage_layer_kernel(const float* __restrict__ agg, const float* __restrict__ cnt,
                                  const float* __restrict__ hin,
                                  const float* __restrict__ Wl, const float* __restrict__ bias,
                                  const float* __restrict__ Wr, float* __restrict__ hout) {
  __shared__ float sMean[16 * 132];  // raw agg tile (padded stride vs bank conflicts)
  __shared__ float sRoot[16 * 132];
  __shared__ float sInv[16];

  const int t = threadIdx.x;
  const int node0 = blockIdx.x * 16;

  if (t < 16) sInv[t] = 1.0f / fmaxf(cnt[node0 + t], 1.0f);

  // Async-DMA both 16x128 tiles into LDS (2 x b128 per thread per tile).
  for (int i = t; i < 512; i += 256) {
    int row = i >> 5;
    int c = (i & 31) << 2;
    async_copy_b128(&sMean[row * 132 + c], agg + (size_t)(node0 + row) * CH + c);
    async_copy_b128(&sRoot[row * 132 + c], hin + (size_t)(node0 + row) * CH + c);
  }
  wait_async_all();
  __syncthreads();

  const int w = t >> 5;
  const int lane = t & 31;
  const int half = lane >> 4;   // 0: lanes 0-15, 1: lanes 16-31
  const int l16 = lane & 15;
  const int col0 = w * 16;

  v8f accM = {};
  v8f accR = {};
  for (int k0 = 0; k0 < CH; k0 += 4) {
    const int ka = k0 + (half << 1);  // A VGPR0/1 hold K=ka, ka+1 per half
    v2f am = *(const v2f*)(&sMean[l16 * 132 + ka]);
    v2f ar = *(const v2f*)(&sRoot[l16 * 132 + ka]);
    v2f bl;
    bl.x = Wl[ka * CH + col0 + l16];
    bl.y = Wl[(ka + 1) * CH + col0 + l16];
    v2f br;
    br.x = Wr[ka * CH + col0 + l16];
    br.y = Wr[(ka + 1) * CH + col0 + l16];
    accM = __builtin_amdgcn_wmma_f32_16x16x4_f32(false, am, false, bl, (short)0, accM, false, false);
    accR = __builtin_amdgcn_wmma_f32_16x16x4_f32(false, ar, false, br, (short)0, accR, false, false);
  }

  const int col = col0 + l16;
  const float bv = bias[col];
#pragma unroll
  for (int r = 0; r < 8; r++) {
    int m = r + (half << 3);  // C/D layout: VGPR r -> M=r (lo half) / r+8 (hi half)
    float v = accM[r] * sInv[m] + accR[r] + bv;
    hout[(size_t)(node0 + m) * CH + col] = fmaxf(v, 0.0f);
  }
}

// ---------------------------------------------------------------------------
// Decoder: out = alpha * (h @ Wd + bd) + (1-alpha) * x
// ---------------------------------------------------------------------------
__launch_bounds__(256, 2)
__global__ void decoder_kernel(const float* __restrict__ h, const float* __restrict__ Wd,
                               const float* __restrict__ bd, const float* __restrict__ x,
                               const float* __restrict__ alpha_p, float* __restrict__ out) {
  __shared__ float sA[16 * 132];
  const int t = threadIdx.x;
  const int node0 = blockIdx.x * 16;

  for (int i = t; i < 512; i += 256) {
    int row = i >> 5;
    int c = (i & 31) << 2;
    async_copy_b128(&sA[row * 132 + c], h + (size_t)(node0 + row) * CH + c);
  }
  wait_async_all();
  __syncthreads();

  const int w = t >> 5;
  const int lane = t & 31;
  const int half = lane >> 4;
  const int l16 = lane & 15;
  const int col0 = w * 16;

  v8f acc = {};
  for (int k0 = 0; k0 < CH; k0 += 4) {
    const int ka = k0 + (half << 1);
    v2f a = *(const v2f*)(&sA[l16 * 132 + ka]);
    v2f b;
    b.x = Wd[ka * CH + col0 + l16];
    b.y = Wd[(ka + 1) * CH + col0 + l16];
    acc = __builtin_amdgcn_wmma_f32_16x16x4_f32(false, a, false, b, (short)0, acc, false, false);
  }

  const float alpha = *alpha_p;
  const float beta = 1.0f - alpha;
  const int col = col0 + l16;
  const float bv = bd[col];
#pragma unroll
  for (int r = 0; r < 8; r++) {
    int m = r + (half << 3);
    size_t o = (size_t)(node0 + m) * CH + col;
    out[o] = alpha * (acc[r] + bv) + beta * x[o];
  }
}

// ---------------------------------------------------------------------------
extern "C" void kernel_launch(void* const* d_in, const int* in_sizes, int n_in,
                              void* d_out, int out_size, void* d_ws, size_t ws_size,
                              hipStream_t stream) {
  const float* x   = (const float*)d_in[0];
  const int*   ei  = (const int*)d_in[1];
  const float* W1l = (const float*)d_in[2];
  const float* b1  = (const float*)d_in[3];
  const float* W1r = (const float*)d_in[4];
  const float* W2l = (const float*)d_in[5];
  const float* b2  = (const float*)d_in[6];
  const float* W2r = (const float*)d_in[7];
  const float* Wd  = (const float*)d_in[8];
  const float* bd  = (const float*)d_in[9];
  const float* alpha = (const float*)d_in[10];

  const int* src = ei;            // edge_index[0, :]
  const int* dst = ei + N_EDGES;  // edge_index[1, :]

  float* ws  = (float*)d_ws;
  float* agg = ws;                          // N*CH
  float* cnt = agg + (size_t)N_NODES * CH;  // N
  float* h1  = cnt + N_NODES;               // N*CH
  float* h2  = h1 + (size_t)N_NODES * CH;   // N*CH
  float* outp = (float*)d_out;

  const int gemm_blocks = N_NODES / 16;   // 6250
  const int scat_blocks = N_EDGES / 8;    // 200000 (8 edges/block * 32 lanes)
  const int cnt_blocks  = N_EDGES / 256;  // 6250

  // Layer 1
  zero_kernel<<<2048, 256, 0, stream>>>((float4*)agg, (N_NODES * CH + N_NODES) / 4);
  count_kernel<<<cnt_blocks, 256, 0, stream>>>(dst, cnt);
  scatter_kernel<<<scat_blocks, 256, 0, stream>>>(x, src, dst, agg);
  sage_layer_kernel<<<gemm_blocks, 256, 0, stream>>>(agg, cnt, x, W1l, b1, W1r, h1);

  // Layer 2 (counts reused)
  zero_kernel<<<2048, 256, 0, stream>>>((float4*)agg, (N_NODES * CH) / 4);
  scatter_kernel<<<scat_blocks, 256, 0, stream>>>(h1, src, dst, agg);
  sage_layer_kernel<<<gemm_blocks, 256, 0, stream>>>(agg, cnt, h1, W2l, b2, W2r, h2);

  // Decoder + residual blend
  decoder_kernel<<<gemm_blocks, 256, 0, stream>>>(h2, Wd, bd, x, alpha, outp);
}